// SimplifiedGCN_44959717654591
// MI455X (gfx1250) — compile-verified
//
#include <hip/hip_runtime.h>

#define DIM 128

typedef __attribute__((ext_vector_type(2))) float v2f;
typedef __attribute__((ext_vector_type(8))) float v8f;

// ---------------------------------------------------------------------------
// deg[i] = 1.0 (self-loop contributes 1 to every node's in-degree)
// ---------------------------------------------------------------------------
__global__ void init_deg_kernel(float* __restrict__ deg, int n) {
    int i = blockIdx.x * blockDim.x + threadIdx.x;
    if (i < n) deg[i] = 1.0f;
}

// deg[col[e]] += 1 for every edge
__global__ void degree_kernel(const int* __restrict__ col, float* __restrict__ deg, int E) {
    int e = blockIdx.x * blockDim.x + threadIdx.x;
    if (e < E) atomicAdd(&deg[col[e]], 1.0f);
}

// deg[i] = 1/sqrt(deg[i])   (deg >= 1 always, so no zero guard needed)
__global__ void rsqrt_kernel(float* __restrict__ deg, int n) {
    int i = blockIdx.x * blockDim.x + threadIdx.x;
    if (i < n) deg[i] = rsqrtf(deg[i]);
}

// ---------------------------------------------------------------------------
// Self-loop term: h_out[i,:] = dis[i]^2 * h_in[i,:]
// Plain stores: runs before the edge scatter (stream order), so it both adds
// the self-loop message AND initializes h_out — no separate zeroing pass,
// and every element is overwritten each call (graph-replay deterministic).
// One wave32 per node, one float4 per lane (32*4 = 128 = DIM).
// ---------------------------------------------------------------------------
__global__ void selfloop_kernel(const float* __restrict__ hin, float* __restrict__ hout,
                                const float* __restrict__ dis, int n) {
    int gid  = blockIdx.x * blockDim.x + threadIdx.x;
    int node = gid >> 5;
    int lane = gid & 31;
    if (node >= n) return;
    float d = dis[node];
    float w = d * d;
    size_t off = (size_t)node * DIM + (size_t)lane * 4;
    float4 v = *(const float4*)(hin + off);
    float4 o = make_float4(w * v.x, w * v.y, w * v.z, w * v.w);
    *(float4*)(hout + off) = o;
}

// ---------------------------------------------------------------------------
// Edge scatter: h_out[row[e],:] += dis[row]*dis[col] * h_in[col[e],:]
// One wave32 per edge: 512-byte coalesced gather (global_load_b128 per lane),
// 4 global_atomic_add_f32 per lane for the scatter.
// ---------------------------------------------------------------------------
__global__ void propagate_kernel(const float* __restrict__ hin, float* __restrict__ hout,
                                 const int* __restrict__ row, const int* __restrict__ col,
                                 const float* __restrict__ dis, int E) {
    int gid  = blockIdx.x * blockDim.x + threadIdx.x;
    int e    = gid >> 5;
    int lane = gid & 31;
    if (e >= E) return;
    int r = row[e];
    int c = col[e];
    float w = dis[r] * dis[c];
    float4 v = *(const float4*)(hin + (size_t)c * DIM + (size_t)lane * 4);
    float* dst = hout + (size_t)r * DIM + (size_t)lane * 4;
    atomicAdd(dst + 0, w * v.x);
    atomicAdd(dst + 1, w * v.y);
    atomicAdd(dst + 2, w * v.z);
    atomicAdd(dst + 3, w * v.w);
}

// ---------------------------------------------------------------------------
// out = A @ W^T + b  via V_WMMA_F32_16X16X4_F32 (exact fp32, matches reference).
// A: [M,128] (M = 100000 = 6250*16), W: [128,128], out: [M,128].
// Block = 256 threads = 8 waves; block b owns M-tile rows [16b,16b+16);
// wave w owns N-tile cols [16w,16w+16). K loop: 128 in steps of 4 -> 32 WMMAs.
//
// A fragment (16x4 f32, ISA layout): lanes 0-15 hold A[M=lane][k0,k0+1],
//   lanes 16-31 hold A[M=lane-16][k0+2,k0+3]  -> one 8B v2f load.
// B fragment (4x16, B[k][n] = W[n][k]): lanes 0-15 hold K={k0,k0+1},
//   lanes 16-31 hold K={k0+2,k0+3}, N = lane&15 -> one 8B v2f load.
// C/D (16x16 f32): VGPR r <-> row m0 + 8*(lane>=16) + r, col n0 + (lane&15).
// No tails anywhere -> EXEC all-1s around WMMA as required.
// ---------------------------------------------------------------------------
__global__ void wmma_gemm_kernel(const float* __restrict__ A,
                                 const float* __restrict__ W,
                                 const float* __restrict__ bias,
                                 float* __restrict__ out) {
    const int lane = threadIdx.x & 31;
    const int wave = threadIdx.x >> 5;          // 0..7 -> N tile
    const int m0   = blockIdx.x * 16;
    const int n0   = wave * 16;

    const int half  = lane >> 4;                // 0: lanes 0-15, 1: lanes 16-31
    const int arow  = lane & 15;                // M within tile (A frag)
    const int koff  = half * 2;                 // K sub-offset for this lane half
    const int ncol  = lane & 15;                // N within tile (B frag)

    const float* Arow = A + (size_t)(m0 + arow) * DIM;
    const float* Wrow = W + (size_t)(n0 + ncol) * DIM;   // B[k][n] = W[n][k]

    v8f c = {};                                 // fp32 accumulator, starts at 0
#pragma unroll
    for (int k0 = 0; k0 < DIM; k0 += 4) {
        v2f a = *(const v2f*)(Arow + k0 + koff);
        v2f b = *(const v2f*)(Wrow + k0 + koff);
        // 8 args: (neg_a, A, neg_b, B, c_mod, C, reuse_a, reuse_b)
        c = __builtin_amdgcn_wmma_f32_16x16x4_f32(
                /*neg_a=*/false, a, /*neg_b=*/false, b,
                /*c_mod=*/(short)0, c, /*reuse_a=*/false, /*reuse_b=*/false);
    }

    const float bv = bias[n0 + ncol];
    float* orow = out + (size_t)(m0 + half * 8) * DIM + n0 + ncol;
#pragma unroll
    for (int r = 0; r < 8; ++r)
        orow[(size_t)r * DIM] = c[r] + bv;
}

// ---------------------------------------------------------------------------
extern "C" void kernel_launch(void* const* d_in, const int* in_sizes, int n_in,
                              void* d_out, int out_size, void* d_ws, size_t ws_size,
                              hipStream_t stream) {
    const float* x     = (const float*)d_in[0];      // [N, 128]
    const int*   ei    = (const int*)  d_in[1];      // [2, E]
    const float* lin_w = (const float*)d_in[2];      // [128, 128]
    const float* lin_b = (const float*)d_in[3];      // [128]
    float*       out   = (float*)d_out;              // [N, 128]

    const int N = in_sizes[0] / DIM;                 // 100000
    const int E = in_sizes[1] / 2;                   // 1600000
    const int* row = ei;                             // ei[0]
    const int* col = ei + E;                         // ei[1]

    // Workspace layout (all 16B-aligned): deg | h1 | h2
    float* deg = (float*)d_ws;                       // N floats
    float* h1  = deg + N;                            // N*DIM floats
    float* h2  = h1 + (size_t)N * DIM;               // N*DIM floats

    const int T = 256;
    const long long nodeThreads = (long long)N * 32;
    const long long edgeThreads = (long long)E * 32;

    // 1) normalization weights
    init_deg_kernel<<<(N + T - 1) / T, T, 0, stream>>>(deg, N);
    degree_kernel  <<<(E + T - 1) / T, T, 0, stream>>>(col, deg, E);
    rsqrt_kernel   <<<(N + T - 1) / T, T, 0, stream>>>(deg, N);

    // 2) layer 1: x -> h1   (self-loop store initializes, then edge scatter-add)
    selfloop_kernel <<<(nodeThreads + T - 1) / T, T, 0, stream>>>(x, h1, deg, N);
    propagate_kernel<<<(edgeThreads + T - 1) / T, T, 0, stream>>>(x, h1, row, col, deg, E);

    // 3) layer 2: h1 -> h2
    selfloop_kernel <<<(nodeThreads + T - 1) / T, T, 0, stream>>>(h1, h2, deg, N);
    propagate_kernel<<<(edgeThreads + T - 1) / T, T, 0, stream>>>(h1, h2, row, col, deg, E);

    // 4) dense head: out = h2 @ W^T + b  (fp32 WMMA, 6250 M-tiles x 8 N-tiles)
    wmma_gemm_kernel<<<N / 16, T, 0, stream>>>(h2, lin_w, lin_b, out);
}